// SketchRenderNet_67783173865975
// MI455X (gfx1250) — compile-verified
//
#include <hip/hip_runtime.h>

typedef __attribute__((ext_vector_type(2))) float v2f;
typedef __attribute__((ext_vector_type(8))) float v8f;

#define IMSIZE 512
#define NSEG   128
#define SMOOTH 8.0f
#define EPSF   1e-10f

// One wave32 renders one 16-pixel strip of a row.
// WMMA tile: A[16 segs x 4] * B[4 x 16 pixels] -> n = a1*yp + dx2*xp + c0.
// C/D layout (f32 16x16): lane l, vgpr r -> pixel N = l%16, segment M = r + 8*(l/16).
__global__ __launch_bounds__(256) void sketch_render_kernel(
    const float* __restrict__ xprm,
    const float* __restrict__ yprm,
    float* __restrict__ out) {

  // Per-segment constants (shared by all 8 waves of the block)
  __shared__ float s_a1[NSEG];
  __shared__ float s_dx2[NSEG];
  __shared__ float s_c0[NSEG];
  // {inv_den, slope, intercept, vert_flag, xlo, xhi, ylo, yhi}
  __shared__ __align__(16) float s_c[NSEG][8];
  // Per-wave distance cache: [wave][k][lane] (lane stride 1 -> conflict-free)
  __shared__ float s_d[8][64][32];

  const int tid = threadIdx.x;

  if (tid < NSEG) {
    const int s = tid;
    const float x0 = xprm[s],     y0 = yprm[s];
    const float x1 = xprm[s + 1], y1 = yprm[s + 1];
    const float dx = x1 - x0;
    const float dy = y1 - y0;
    const float dx2 = dx * dx;                 // (x0-x1)^2
    const float cross = y0 * x1 - x0 * y1;
    s_a1[s]  = dx * dy;                        // (x1-x0)*(y1-y0)
    s_dx2[s] = dx2;
    s_c0[s]  = -dy * cross;                    // (y0-y1)*(y0*x1-x0*y1)
    const float den = dx2 + dy * dy;
    s_c[s][0] = 1.0f / (den + EPSF);
    s_c[s][1] = dy / (dx + EPSF);              // slope
    s_c[s][2] = cross / (dx + EPSF);           // intercept
    s_c[s][3] = (x1 == x0) ? 1.0f : 0.0f;      // vertical-segment flag
    s_c[s][4] = fminf(x0, x1);
    s_c[s][5] = fmaxf(x0, x1);
    s_c[s][6] = fminf(y0, y1);
    s_c[s][7] = fmaxf(y0, y1);
  }
  __syncthreads();

  const int lane  = tid & 31;
  const int n     = lane & 15;                 // pixel column within strip (WMMA N)
  const int h     = lane >> 4;                 // half-wave: segment M offset 8*h
  const int wv    = tid >> 5;                  // wave index in block
  const int strip = blockIdx.x * 8 + wv;       // 16384 strips total
  const int row   = strip >> 5;                // 32 strips per row
  const int cb    = (strip & 31) << 4;

  const float yp = (float)row;
  const float xp = (float)(cb + n);

  // B[4x16] rows: K0=yp, K1=xp, K2=1, K3=0.
  // Layout: lanes 0-15 -> {K0,K1}; lanes 16-31 -> {K2,K3}.
  v2f Bop;
  Bop[0] = h ? 1.0f : yp;
  Bop[1] = h ? 0.0f : xp;

  float dmax = 0.0f;

  // Pass 1: distances for all 128 segments; cache in LDS; track per-lane max.
  // Keep this loop rolled so register pressure stays < 256 VGPRs.
#pragma unroll 1
  for (int g = 0; g < 8; ++g) {
    // A layout (f32 16x4): lanes 0-15 -> {K0=a1, K1=dx2}; lanes 16-31 -> {K2=c0, K3=0}.
    const int sa = 16 * g + n;
    v2f Aop;
    Aop[0] = h ? s_c0[sa] : s_a1[sa];
    Aop[1] = h ? 0.0f     : s_dx2[sa];

    v8f acc = {};
    acc = __builtin_amdgcn_wmma_f32_16x16x4_f32(
        /*neg_a=*/false, Aop, /*neg_b=*/false, Bop,
        /*c_mod=*/(short)0, acc, /*reuse_a=*/false, /*reuse_b=*/false);

#pragma unroll
    for (int r = 0; r < 8; ++r) {
      const int seg = 16 * g + r + 8 * h;      // D element M
      const float4 k0 = *reinterpret_cast<const float4*>(&s_c[seg][0]);
      const float4 k1 = *reinterpret_cast<const float4*>(&s_c[seg][4]);
      float cx = acc[r] * k0.x;                          // n / (den+eps)
      float cy = k0.y * cx + k0.z;                       // slope*cx + intercept
      cy = (k0.w != 0.0f) ? yp : cy;                     // vertical: cy = yp
      cx = __builtin_amdgcn_fmed3f(cx, k1.x, k1.y);      // clamp to bbox (v_med3)
      cy = __builtin_amdgcn_fmed3f(cy, k1.z, k1.w);
      const float ex = cx - xp;
      const float ey = cy - yp;
      const float d = sqrtf(ex * ex + ey * ey + EPSF);
      s_d[wv][g * 8 + r][lane] = d;
      dmax = fmaxf(dmax, d);
    }
  }

  // Per-pixel max over all 128 segments: merge the two half-wave halves.
  dmax = fmaxf(dmax, __shfl_xor(dmax, 16, 32));

  // Pass 2: exponential smooth-min (reference subtracts the MAX; overflows to
  // +inf in fp32 exactly like the reference does).
  float bsum = 0.0f;
#pragma unroll 8
  for (int k = 0; k < 64; ++k)
    bsum += __expf(SMOOTH * (dmax - s_d[wv][k][lane]));
  bsum += __shfl_xor(bsum, 16, 32);

  if (h == 0) {
    const float Cv = (SMOOTH * dmax - __logf(bsum)) * (1.0f / SMOOTH);
    out[row * IMSIZE + (cb + n)] = Cv;
  }
}

extern "C" void kernel_launch(void* const* d_in, const int* in_sizes, int n_in,
                              void* d_out, int out_size, void* d_ws, size_t ws_size,
                              hipStream_t stream) {
  const float* xprm = (const float*)d_in[0];
  const float* yprm = (const float*)d_in[1];
  float* out = (float*)d_out;
  (void)in_sizes; (void)n_in; (void)out_size; (void)d_ws; (void)ws_size;

  // 16384 strips (512 rows x 32 strips), one wave32 per strip,
  // 8 waves (256 threads) per block -> 2048 blocks.
  dim3 grid((IMSIZE * (IMSIZE / 16)) / 8);
  dim3 block(256);
  sketch_render_kernel<<<grid, block, 0, stream>>>(xprm, yprm, out);
}